// DeepCross_43224550867491
// MI455X (gfx1250) — compile-verified
//
#include <hip/hip_runtime.h>
#include <hip/hip_bf16.h>
#include <math.h>

// Problem constants (match reference)
#define BB   8192     // batch
#define TT   128      // trees
#define EE   64       // leaf emb dim
#define DD   (TT*EE)  // 8192
#define LL   4
#define H1   32
#define H2   16

// Tiling
#define MT   16       // rows per workgroup
#define KC   512      // K chunk (8 trees)
#define NCH  (DD/KC)  // 16 chunks
#define XS   516      // padded LDS row stride (floats): conflict-free, 16B aligned

typedef float  v2f __attribute__((ext_vector_type(2)));
typedef float  v8f __attribute__((ext_vector_type(8)));

__global__ __launch_bounds__(256, 2)
void deepcross_fused(const int*   __restrict__ x,      // [B,T]
                     const float* __restrict__ emb,    // [V,E]
                     const float* __restrict__ cw,     // [L,D]
                     const float* __restrict__ cb,     // [L]
                     const float* __restrict__ w1,     // [D,H1]
                     const float* __restrict__ b1,     // [H1]
                     const float* __restrict__ w2,     // [H1,H2]
                     const float* __restrict__ b2,     // [H2]
                     const float* __restrict__ wf,     // [H2+D]
                     const float* __restrict__ bf,     // [1]
                     float*       __restrict__ out)    // [B]
{
    __shared__ float lds_x0[MT * XS];          // 33024 B : x0 chunk [16][512] (padded)
    __shared__ float cacc[8][MT][H1];          // 16384 B : per-wave WMMA partials
    __shared__ float rowacc[MT][5];            //   320 B : g0..g3, wide
    __shared__ float c4s[MT];                  //    64 B : cross scalar per row
    __shared__ float h1s[MT * H1];             //  2048 B
    __shared__ float h2s[MT * H2];             //  1024 B

    const int tid  = threadIdx.x;
    const int lane = tid & 31;
    const int wv   = tid >> 5;                 // wave id 0..7
    const int row0 = blockIdx.x * MT;          // global row base

    // per-thread dot accumulators: 16 threads per row
    const int dr = tid >> 4;                   // row 0..15
    const int dj = tid & 15;                   // lane-in-row

    float g0 = 0.f, g1 = 0.f, g2 = 0.f, g3 = 0.f, wd = 0.f;
    v8f acc0 = {};                             // WMMA C, N-tile [0,16)
    v8f acc1 = {};                             // WMMA C, N-tile [16,32)

    const int m_a  = lane & 15;                // A/B lane -> M (rows), B lane -> N
    const int kh   = (lane >> 4) << 1;         // 0 or 2 (K half select)
    const float* wfw = wf + H2;                // wide part of wf

    for (int chunk = 0; chunk < NCH; ++chunk) {
        __syncthreads();  // previous chunk fully consumed

        // ---- gather 16 rows x 8 trees of embeddings into LDS ----
        {
            const int vec  = tid >> 1;         // 0..127  (row, tree_local)
            const int half = tid & 1;          // which 32 floats of the 64
            const int gr   = vec >> 3;         // row 0..15
            const int tl   = vec & 7;          // tree-in-chunk
            const int tree = chunk * 8 + tl;
            const int idx  = x[(row0 + gr) * TT + tree];
            const float4* src = (const float4*)(emb) + (size_t)idx * (EE / 4) + half * 8;
            float4* dst = (float4*)(&lds_x0[gr * XS + tl * EE + half * 32]);
            #pragma unroll
            for (int i = 0; i < 8; ++i) dst[i] = src[i];
        }

        // prefetch next chunk of w1 into L2/L0 (global_prefetch_b8)
        if (chunk + 1 < NCH) {
            const float* p = w1 + (size_t)(chunk + 1) * KC * H1 + tid * 64;
            __builtin_prefetch(p, 0, 1);
        }

        __syncthreads();

        // ---- cross / wide dot products: 16 threads per row, stride 16 ----
        {
            const float* xr = &lds_x0[dr * XS];
            const int    kg0 = chunk * KC;
            #pragma unroll 4
            for (int k = dj; k < KC; k += 16) {
                const float v  = xr[k];
                const int   kg = kg0 + k;
                g0 += v * cw[0 * DD + kg];
                g1 += v * cw[1 * DD + kg];
                g2 += v * cw[2 * DD + kg];
                g3 += v * cw[3 * DD + kg];
                wd += v * wfw[kg];
            }
        }

        // ---- WMMA: x0_chunk[16 x 512] @ w1_chunk[512 x 32], wave-split K ----
        {
            const int klb = wv * 64;                      // this wave's local K window
            #pragma unroll 4
            for (int s = 0; s < 16; ++s) {
                const int kl = klb + s * 4;               // local K step base
                const int kg = chunk * KC + kl;           // global K
                // A: 16x4 f32 tile from LDS (lane m_a = M row; kh selects K pair)
                const v2f a = *(const v2f*)(&lds_x0[m_a * XS + kl + kh]);
                // B: 4x16 f32 tiles from w1 (row-major [D,32])
                const float* bp = w1 + (size_t)(kg + kh) * H1 + m_a;
                v2f b0; b0.x = bp[0];   b0.y = bp[H1];        // N 0..15
                v2f b1v; b1v.x = bp[16]; b1v.y = bp[H1 + 16]; // N 16..31
                acc0 = __builtin_amdgcn_wmma_f32_16x16x4_f32(
                           false, a, false, b0,  (short)0, acc0, false, false);
                acc1 = __builtin_amdgcn_wmma_f32_16x16x4_f32(
                           false, a, false, b1v, (short)0, acc1, false, false);
            }
        }
    }

    // ---- reduce dot products within each 16-lane half (wave32 shuffles) ----
    #pragma unroll
    for (int msk = 8; msk >= 1; msk >>= 1) {
        g0 += __shfl_xor(g0, msk);
        g1 += __shfl_xor(g1, msk);
        g2 += __shfl_xor(g2, msk);
        g3 += __shfl_xor(g3, msk);
        wd += __shfl_xor(wd, msk);
    }
    if (dj == 0) {
        rowacc[dr][0] = g0; rowacc[dr][1] = g1;
        rowacc[dr][2] = g2; rowacc[dr][3] = g3;
        rowacc[dr][4] = wd;
    }

    // ---- spill per-wave WMMA C tiles (ISA C/D layout: M = v + 8*(lane>=16)) ----
    {
        const int mbase = (lane >> 4) << 3;
        #pragma unroll
        for (int v = 0; v < 8; ++v) {
            cacc[wv][mbase + v][m_a]      = acc0[v];
            cacc[wv][mbase + v][m_a + 16] = acc1[v];
        }
    }
    __syncthreads();

    // ---- cross-layer scalar recurrence per row ----
    if (tid < MT) {
        const float q0 = rowacc[tid][0], q1 = rowacc[tid][1];
        const float q2 = rowacc[tid][2], q3 = rowacc[tid][3];
        float c = 1.0f;
        float s = c * q0 + cb[0]; c = c + s;    // layer 0 (even): temp + cross
        s = c * q1 + cb[1];       c = s;        // layer 1 (odd):  cross
        s = c * q2 + cb[2];       c = c + s;    // layer 2
        s = c * q3 + cb[3];       c = s;        // layer 3
        c4s[tid] = c;
    }
    __syncthreads();

    // ---- h1 = relu(c * (x0@w1) + b1): sum 8 wave partials ----
    #pragma unroll
    for (int e = tid; e < MT * H1; e += 256) {
        const int rr = e >> 5, nn = e & 31;
        float y = 0.f;
        #pragma unroll
        for (int w = 0; w < 8; ++w) y += cacc[w][rr][nn];
        y = c4s[rr] * y + b1[nn];
        h1s[rr * H1 + nn] = y > 0.f ? y : 0.f;
    }
    __syncthreads();

    // ---- h2 = relu(h1 @ w2 + b2): thread (rr, mm) ----
    {
        const int rr = tid >> 4, mm = tid & 15;
        float a = b2[mm];
        #pragma unroll
        for (int n = 0; n < H1; ++n) a += h1s[rr * H1 + n] * w2[n * H2 + mm];
        h2s[rr * H2 + mm] = a > 0.f ? a : 0.f;
    }
    __syncthreads();

    // ---- out = sigmoid(<h2, wf[:16]> + <x0, wf[16:]> + bf) ----
    if (tid < MT) {
        float a = bf[0] + rowacc[tid][4];
        #pragma unroll
        for (int m = 0; m < H2; ++m) a += h2s[tid * H2 + m] * wf[m];
        out[row0 + tid] = 1.0f / (1.0f + __expf(-a));
    }
}

extern "C" void kernel_launch(void* const* d_in, const int* in_sizes, int n_in,
                              void* d_out, int out_size, void* d_ws, size_t ws_size,
                              hipStream_t stream) {
    const int*   x    = (const int*)  d_in[0];
    const float* emb  = (const float*)d_in[1];
    const float* cw   = (const float*)d_in[2];
    const float* cb   = (const float*)d_in[3];
    const float* w1   = (const float*)d_in[4];
    const float* b1   = (const float*)d_in[5];
    const float* w2   = (const float*)d_in[6];
    const float* b2   = (const float*)d_in[7];
    const float* wf   = (const float*)d_in[8];
    const float* bf   = (const float*)d_in[9];
    float*       out  = (float*)d_out;

    dim3 grid(BB / MT);   // 512 workgroups
    dim3 block(256);      // 8 waves (wave32)
    deepcross_fused<<<grid, block, 0, stream>>>(x, emb, cw, cb, w1, b1, w2, b2, wf, bf, out);
}